// Voxelise_74895639708038
// MI455X (gfx1250) — compile-verified
//
#include <hip/hip_runtime.h>

// CDNA5 / gfx1250 Gaussian-KDE voxeliser.
//
// out[g] = 1/(2*pi*b^2) * sum_n exp(-||p_n - g||^2 / (2 b^2))
//
// Strategy: V_WMMA_F32_16X16X4_F32 computes the *entire* exp2 argument:
//   A (16 points x K=4)  : (x, y, z, ||p||^2)
//   B (K=4 x 16 voxels)  : (-2k*gx, -2k*gy, -2k*gz, k),  k = -log2(e)/(2 b^2)
//   C (16x16 splat/lane) : k*||g||^2
//   => D[m][n] = k * ||p_m - g_n||^2   (already scaled for v_exp_f32)
// Epilogue per element: one v_exp_f32 + one v_add_f32, accumulated into a
// per-tile v8f so all 8 exp results per WMMA are independent (no trans-pipe
// serialization, no v_nop hazard fills).

typedef __attribute__((ext_vector_type(2))) float v2f;
typedef __attribute__((ext_vector_type(8))) float v8f;

#define TILES_PER_WAVE 4              // 64 voxels per wave
#define WAVES_PER_BLOCK 8             // 256 threads (wave32)

__global__ __launch_bounds__(256) void voxelise_kde_wmma(
    const float* __restrict__ pc,     // (N, 3) pointcloud
    const float* __restrict__ tp,     // (3, G) target points
    float* __restrict__ out,          // (G,)
    int n_points, int n_vox)
{
    // k = -log2(e) / (2*b*b), b = 10  ->  exp(-d2/(2b^2)) == exp2(k*d2)
    const float kExp  = -0.0072134752044448172f;
    const float kM2   = -2.0f * kExp;               // +0.014426950...
    const float kNorm =  0.0015915494309189534f;    // 1/(2*pi*b^2)

    const int lane = threadIdx.x & 31;
    const int nsub = lane & 15;                     // column n within 16-wide tile
    const bool hi  = lane >= 16;
    const int wave = blockIdx.x * WAVES_PER_BLOCK + (threadIdx.x >> 5);
    const int vbase = wave * (16 * TILES_PER_WAVE);

    // ---- Per-wave voxel setup: B operands and C splats (k*||g||^2) ----
    // B 4x16 f32 layout (mirrors A 16x4, ISA 7.12.2):
    //   VGPR0: lanes 0-15 -> row K=0, lanes 16-31 -> row K=2
    //   VGPR1: lanes 0-15 -> row K=1, lanes 16-31 -> row K=3
    v2f   bop[TILES_PER_WAVE];
    float cval[TILES_PER_WAVE];
#pragma unroll
    for (int t = 0; t < TILES_PER_WAVE; ++t) {
        int v = vbase + t * 16 + nsub;
        if (v > n_vox - 1) v = n_vox - 1;           // clamp (no EXEC masking: WMMA needs EXEC all-1)
        float gx = tp[0 * n_vox + v];
        float gy = tp[1 * n_vox + v];
        float gz = tp[2 * n_vox + v];
        cval[t]  = kExp * (gx * gx + gy * gy + gz * gz);
        bop[t].x = hi ? (kM2 * gz) : (kM2 * gx);    // rows K=2 / K=0
        bop[t].y = hi ? kExp       : (kM2 * gy);    // rows K=3 / K=1
    }

    // 8 independent accumulator chains per tile -> exp results never feed a
    // common serial chain inside the loop.
    v8f vacc[TILES_PER_WAVE];
#pragma unroll
    for (int t = 0; t < TILES_PER_WAVE; ++t)
        vacc[t] = (v8f){0.f, 0.f, 0.f, 0.f, 0.f, 0.f, 0.f, 0.f};

    // ---- Main loop: 16 points per iteration, 4 WMMAs sharing one A operand.
    // Unroll x2 so the next b96 load overlaps current WMMA + exp work. ----
#pragma unroll 2
    for (int p0 = 0; p0 < n_points; p0 += 16) {
        const float* pp = pc + 3 * (p0 + nsub);     // point m = nsub of this tile
        float x = pp[0];
        float y = pp[1];
        float z = pp[2];
        float pn2 = x * x + y * y + z * z;

        // A 16x4 f32 layout (ISA 7.12.2):
        //   VGPR0: lanes 0-15 -> K=0 (x), lanes 16-31 -> K=2 (z)
        //   VGPR1: lanes 0-15 -> K=1 (y), lanes 16-31 -> K=3 (||p||^2)
        v2f a;
        a.x = hi ? z   : x;
        a.y = hi ? pn2 : y;

#pragma unroll
        for (int t = 0; t < TILES_PER_WAVE; ++t) {
            float g = cval[t];
            v8f c = {g, g, g, g, g, g, g, g};
            // 8 args: (neg_a, A, neg_b, B, c_mod, C, reuse_a, reuse_b)
            v8f d = __builtin_amdgcn_wmma_f32_16x16x4_f32(
                false, a, false, bop[t], (short)0, c, false, false);
#pragma unroll
            for (int i = 0; i < 8; ++i)
                vacc[t][i] += __builtin_amdgcn_exp2f(d[i]);   // v_exp_f32
        }
    }

    // ---- Epilogue: horizontal sum of 8 chains, then fold the two lane-halves
    // (rows m are split across lanes 0-15 / 16-31 in the C/D layout). ----
#pragma unroll
    for (int t = 0; t < TILES_PER_WAVE; ++t) {
        v8f va = vacc[t];
        float s01 = va[0] + va[1];
        float s23 = va[2] + va[3];
        float s45 = va[4] + va[5];
        float s67 = va[6] + va[7];
        float acc = (s01 + s23) + (s45 + s67);
        float total = acc + __shfl_xor(acc, 16, 32);
        int v = vbase + t * 16 + nsub;              // unclamped
        if (!hi && v < n_vox)
            out[v] = kNorm * total;
    }
}

extern "C" void kernel_launch(void* const* d_in, const int* in_sizes, int n_in,
                              void* d_out, int out_size, void* d_ws, size_t ws_size,
                              hipStream_t stream) {
    const float* pc = (const float*)d_in[0];        // (N, 3) float32
    const float* tp = (const float*)d_in[1];        // (3, G) float32
    float* out = (float*)d_out;                     // (G,) float32

    int n_points = in_sizes[0] / 3;                 // 2048
    int n_vox    = out_size;                        // 274625

    int vox_per_wave = 16 * TILES_PER_WAVE;         // 64
    int n_waves  = (n_vox + vox_per_wave - 1) / vox_per_wave;
    int n_blocks = (n_waves + WAVES_PER_BLOCK - 1) / WAVES_PER_BLOCK;

    voxelise_kde_wmma<<<n_blocks, WAVES_PER_BLOCK * 32, 0, stream>>>(
        pc, tp, out, n_points, n_vox);
}